// LSTMN_75660143886512
// MI455X (gfx1250) — compile-verified
//
#include <hip/hip_runtime.h>
#include <math.h>

// Problem constants (match reference)
#define BB   8
#define TT   256
#define DIMD 256
#define FEAT 256
#define OUTN 10
#define ATT  10
#define GATE 1024   // 4*DIM
#define NCOL 512    // FEAT+DIM
#define NTOT 2048   // B*T
#define APAD 16     // padded ATT stride

typedef float v2f __attribute__((ext_vector_type(2)));
typedef float v8f __attribute__((ext_vector_type(8)));
typedef int   v4i __attribute__((vector_size(4 * sizeof(int))));  // matches builtin param

#define AS_GLOBAL __attribute__((address_space(1)))
#define AS_LDS    __attribute__((address_space(3)))

#if __has_builtin(__builtin_amdgcn_global_load_async_to_lds_b128) && \
    __has_builtin(__builtin_amdgcn_s_wait_asynccnt)
#define USE_ASYNC_LDS 1
#else
#define USE_ASYNC_LDS 0
#endif

__device__ __forceinline__ float sigf(float x) { return 1.0f / (1.0f + __expf(-x)); }

// ---------------------------------------------------------------------------
// Kernel 1: XW[n,j] = b[j] + sum_f x[n,f] * W[j, DIM+f]   (fp32 WMMA GEMM)
// One wave per 16x16 output tile; K-loop in steps of 4 with v_wmma_f32_16x16x4_f32.
// ---------------------------------------------------------------------------
__global__ __launch_bounds__(256) void xw_gemm_wmma(const float* __restrict__ x,
                                                    const float* __restrict__ W,
                                                    const float* __restrict__ bias,
                                                    float* __restrict__ XW) {
    const int wave = (blockIdx.x * blockDim.x + threadIdx.x) >> 5;  // 0..8191
    const int lane = threadIdx.x & 31;
    const int tn = wave >> 6;          // 0..127  (n tile)
    const int tj = wave & 63;          // 0..63   (j tile)
    const int n0 = tn * 16, j0 = tj * 16;
    const int m  = lane & 15;          // M index for A, N index for B
    const int kh = lane >> 4;          // selects K pair {2*kh, 2*kh+1}

    const float* arow = x + (size_t)(n0 + m) * FEAT + 2 * kh;           // A: x[n0+m][k]
    const float* brow = W + (size_t)(j0 + m) * NCOL + DIMD + 2 * kh;    // B: W[j0+m][DIM+k]
    __builtin_prefetch(arow, 0, 0);
    __builtin_prefetch(brow, 0, 0);
    __builtin_prefetch(arow + 128, 0, 0);
    __builtin_prefetch(brow + 128, 0, 0);

    v8f acc = {};
    #pragma unroll 8
    for (int kb = 0; kb < FEAT; kb += 4) {
        v2f af = *(const v2f*)(arow + kb);
        v2f bf = *(const v2f*)(brow + kb);
        // D = A(16x4,f32) * B(4x16,f32) + C   -> v_wmma_f32_16x16x4_f32
        acc = __builtin_amdgcn_wmma_f32_16x16x4_f32(
            /*neg_a=*/false, af, /*neg_b=*/false, bf,
            /*c_mod=*/(short)0, acc, /*reuse_a=*/false, /*reuse_b=*/false);
    }

    // D layout: vgpr i, lanes 0-15: M=i,N=lane ; lanes 16-31: M=8+i,N=lane-16
    const int ncol  = lane & 15;
    const int mbase = (lane >> 4) * 8;
    const float bj  = bias[j0 + ncol];
    #pragma unroll
    for (int i = 0; i < 8; ++i)
        XW[(size_t)(n0 + mbase + i) * GATE + j0 + ncol] = acc[i] + bj;
}

// ---------------------------------------------------------------------------
// Kernel 2: WxX[n,a] = sum_f x[n,f] * W_x[a,f]   (padded [NTOT][APAD])
// ---------------------------------------------------------------------------
__global__ void wxx_kernel(const float* __restrict__ x, const float* __restrict__ W_x,
                           float* __restrict__ WxX) {
    int idx = blockIdx.x * blockDim.x + threadIdx.x;
    if (idx >= NTOT * APAD) return;
    int n = idx >> 4, a = idx & 15;
    float acc = 0.0f;
    if (a < ATT) {
        const float* xr = x + (size_t)n * FEAT;
        const float* wr = W_x + (size_t)a * FEAT;
        #pragma unroll 4
        for (int f = 0; f < FEAT; ++f) acc = fmaf(xr[f], wr[f], acc);
    }
    WxX[idx] = acc;
}

// ---------------------------------------------------------------------------
// Kernel 3: WhhT[d,j] = W[j,d]  (transpose hidden part of W for coalesced recurrent matvec)
// ---------------------------------------------------------------------------
__global__ void whh_transpose(const float* __restrict__ W, float* __restrict__ WhhT) {
    int idx = blockIdx.x * blockDim.x + threadIdx.x;
    if (idx >= DIMD * GATE) return;
    int d = idx >> 10, j = idx & 1023;
    WhhT[idx] = W[(size_t)j * NCOL + d];
}

// ---------------------------------------------------------------------------
// Kernel 4: sequential LSTMN recurrence. Single workgroup (32 waves), persistent.
// ---------------------------------------------------------------------------
__global__ __launch_bounds__(1024) void lstmn_seq(
    const float* __restrict__ W_h, const float* __restrict__ W_ht,
    const float* __restrict__ v, const float* __restrict__ ht0,
    const float* __restrict__ ct0, const float* __restrict__ proj_w,
    const float* __restrict__ proj_b,
    const float* __restrict__ XW, const float* __restrict__ WxX,
    const float* __restrict__ WhhT, float* __restrict__ PC,
    float* __restrict__ out, float* __restrict__ hs) {

    __shared__ float A_s[TT][APAD];   // cached W_h @ h_k
    __shared__ float Bh_s[TT][APAD];  // cached W_ht @ ht_k
    __shared__ float Wx_s[TT][APAD];  // per-sample x_k @ W_x^T
    __shared__ float p_s[TT];
    __shared__ float ht_s[DIMD];      // carry: ht_new (adaptive hidden)
    __shared__ float ctn_s[DIMD];     // carry: ct_new (adaptive cell) — never clobbered
    __shared__ float h_s[DIMD];       // current step's output hidden
    __shared__ float whx_s[GATE];
    __shared__ float v_s[APAD];
    __shared__ float red[16];

    const int tid  = threadIdx.x;
    const int lane = tid & 31;
    const int wid  = tid >> 5;

    if (tid < DIMD) { ht_s[tid] = ht0[tid]; ctn_s[tid] = ct0[tid]; }
    if (tid < APAD) v_s[tid] = (tid < ATT) ? v[tid] : 0.0f;
    __syncthreads();

    for (int b = 0; b < BB; ++b) {
#if USE_ASYNC_LDS
        // Async DMA this sample's WxX tile (16KB) into LDS (one b128 per thread),
        // overlapped with zeroing the attention caches; fence with s_wait_asynccnt 0.
        {
            AS_GLOBAL v4i* gsrc =
                (AS_GLOBAL v4i*)(const_cast<float*>(WxX) + (size_t)b * TT * APAD + tid * 4);
            AS_LDS v4i* ldst = (AS_LDS v4i*)((float*)Wx_s + tid * 4);
            __builtin_amdgcn_global_load_async_to_lds_b128(gsrc, ldst, 0, 0);
        }
        for (int i = tid; i < TT * APAD; i += 1024) {
            ((float*)A_s)[i]  = 0.0f;
            ((float*)Bh_s)[i] = 0.0f;
        }
        __builtin_amdgcn_s_wait_asynccnt(0);
        __syncthreads();
#else
        for (int i = tid; i < TT * APAD; i += 1024) {
            ((float*)Wx_s)[i] = WxX[(size_t)b * TT * APAD + i];
            ((float*)A_s)[i]  = 0.0f;
            ((float*)Bh_s)[i] = 0.0f;
        }
        __syncthreads();
#endif

        float* Hg = hs + (size_t)b * TT * DIMD;   // hidden history == output slab

        for (int t = 0; t < TT; ++t) {
            if (t > 0) {
                // 1) scores_k = v . tanh(A_k + WxX_k + Bh_k), k < t
                if (tid < t) {
                    float s = 0.0f;
                    #pragma unroll
                    for (int a = 0; a < ATT; ++a)
                        s = fmaf(v_s[a], tanhf(A_s[tid][a] + Wx_s[tid][a] + Bh_s[tid][a]), s);
                    p_s[tid] = s;
                }
                __syncthreads();
                // 2) softmax over k < t (wave 0, wave32 shuffles)
                if (wid == 0) {
                    float mx = -INFINITY;
                    for (int k = lane; k < t; k += 32) mx = fmaxf(mx, p_s[k]);
                    #pragma unroll
                    for (int off = 16; off > 0; off >>= 1)
                        mx = fmaxf(mx, __shfl_down(mx, off, 32));
                    mx = __shfl(mx, 0, 32);
                    float sum = 0.0f;
                    for (int k = lane; k < t; k += 32) {
                        float e = __expf(p_s[k] - mx);
                        p_s[k] = e; sum += e;
                    }
                    #pragma unroll
                    for (int off = 16; off > 0; off >>= 1)
                        sum += __shfl_down(sum, off, 32);
                    sum = __shfl(sum, 0, 32);
                    float inv = 1.0f / sum;
                    for (int k = lane; k < t; k += 32) p_s[k] *= inv;
                }
                __syncthreads();
                // 3) ht_new = p @ H ; ct_new = p @ PC  (coalesced across d)
                if (tid < 2 * DIMD) {
                    const int d = tid & (DIMD - 1);
                    const float* M = (tid < DIMD) ? Hg : PC;
                    float acc = 0.0f;
                    for (int k = 0; k < t; ++k)
                        acc = fmaf(p_s[k], M[(size_t)k * DIMD + d], acc);
                    if (tid < DIMD) ht_s[d] = acc; else ctn_s[d] = acc;
                }
                __syncthreads();
            }

            // 4) whx[j] = XW[n,j] + sum_d WhhT[d,j] * ht_new[d]   (all 1024 threads)
            {
                const int n = b * TT + t;
                float acc = XW[(size_t)n * GATE + tid];
                const float* wt = WhhT + tid;
                #pragma unroll 4
                for (int d = 0; d < DIMD; ++d)
                    acc = fmaf(ht_s[d], wt[(size_t)d * GATE], acc);
                whx_s[tid] = acc;
            }
            __syncthreads();

            // 5) gates -> h, c ; store into history (global, L2-resident)
            if (tid < DIMD) {
                float f  = sigf(whx_s[tid]);
                float o  = sigf(whx_s[DIMD + tid]);
                float ii = sigf(whx_s[2 * DIMD + tid]);
                float ch = tanhf(whx_s[3 * DIMD + tid]);
                float c  = f * ctn_s[tid] + ii * ch;
                float h  = o * tanhf(c);
                h_s[tid] = h;
                Hg[(size_t)t * DIMD + tid] = h;
                PC[(size_t)t * DIMD + tid] = c;
            }
            __threadfence();
            __syncthreads();

            // 6) cache A[t] = W_h @ h, Bh[t] = W_ht @ ht_new (20 waves)
            if (wid < 20) {
                const int a = (wid < 10) ? wid : (wid - 10);
                const bool isA = (wid < 10);
                const float* wr  = isA ? (W_h + (size_t)a * DIMD) : (W_ht + (size_t)a * DIMD);
                const float* vec = isA ? h_s : ht_s;
                float acc = 0.0f;
                for (int d = lane; d < DIMD; d += 32) acc = fmaf(wr[d], vec[d], acc);
                #pragma unroll
                for (int off = 16; off > 0; off >>= 1) acc += __shfl_down(acc, off, 32);
                if (lane == 0) { if (isA) A_s[t][a] = acc; else Bh_s[t][a] = acc; }
            }
            // 7) final projection at last step (waves 20..29)
            if (t == TT - 1 && wid >= 20 && wid < 30) {
                const int a = wid - 20;
                float acc = 0.0f;
                for (int d = lane; d < DIMD; d += 32)
                    acc = fmaf(proj_w[(size_t)a * DIMD + d], h_s[d], acc);
                #pragma unroll
                for (int off = 16; off > 0; off >>= 1) acc += __shfl_down(acc, off, 32);
                if (lane == 0) red[a] = acc + proj_b[a];
            }
            __syncthreads();
            if (t == TT - 1 && tid == 0) {
                float mx = -INFINITY;
                for (int a = 0; a < OUTN; ++a) mx = fmaxf(mx, red[a]);
                float sum = 0.0f; float e[OUTN];
                for (int a = 0; a < OUTN; ++a) { e[a] = __expf(red[a] - mx); sum += e[a]; }
                float inv = 1.0f / sum;
                for (int a = 0; a < OUTN; ++a) out[b * OUTN + a] = e[a] * inv;
            }
            __syncthreads();
        }
        // Carry across samples: ht_s holds ht_new(T-1), ctn_s holds ct_new(T-1).
        // Both arrays survive untouched into the next sample's t=0.
    }
}

// ---------------------------------------------------------------------------
extern "C" void kernel_launch(void* const* d_in, const int* in_sizes, int n_in,
                              void* d_out, int out_size, void* d_ws, size_t ws_size,
                              hipStream_t stream) {
    const float* x      = (const float*)d_in[0];
    const float* W      = (const float*)d_in[1];
    const float* bias   = (const float*)d_in[2];
    const float* W_h    = (const float*)d_in[3];
    const float* W_x    = (const float*)d_in[4];
    const float* W_ht   = (const float*)d_in[5];
    const float* v      = (const float*)d_in[6];
    const float* ht0    = (const float*)d_in[7];
    const float* ct0    = (const float*)d_in[8];
    const float* proj_w = (const float*)d_in[9];
    const float* proj_b = (const float*)d_in[10];

    float* out = (float*)d_out;            // [B*OUT]
    float* hs  = out + BB * OUTN;          // [B*T*DIM]

    float* ws   = (float*)d_ws;
    float* XW   = ws;                                   // NTOT*GATE
    float* WxX  = XW + (size_t)NTOT * GATE;             // NTOT*APAD
    float* WhhT = WxX + (size_t)NTOT * APAD;            // DIM*GATE
    float* PC   = WhhT + (size_t)DIMD * GATE;           // T*DIM

    // Parallel precompute (all the data-parallel FLOPs, WMMA f32)
    xw_gemm_wmma<<<1024, 256, 0, stream>>>(x, W, bias, XW);
    wxx_kernel<<<(NTOT * APAD + 255) / 256, 256, 0, stream>>>(x, W_x, WxX);
    whh_transpose<<<(DIMD * GATE + 255) / 256, 256, 0, stream>>>(W, WhhT);
    // Sequential recurrence: one persistent workgroup
    lstmn_seq<<<1, 1024, 0, stream>>>(W_h, W_ht, v, ht0, ct0, proj_w, proj_b,
                                      XW, WxX, WhhT, PC, out, hs);
}